// UpsampleTSDConv_40647570489920
// MI455X (gfx1250) — compile-verified
//
#include <hip/hip_runtime.h>

// Problem constants (from reference)
#define H_IN   160
#define W_IN   160
#define RES_H  160
#define RES_W  320
#define OUT_HH 320
#define OUT_WW 640

// Tiling: one block -> 16x64 output tile of one batch image
#define TOH 16
#define TOW 64
// x tile staged in LDS: rows [kr0-2, kr0+9], cols [cl0/2-2, cl0/2+17]
#define XTH 12
#define XTW 20

// res(r,c) of the checkerboard-upsampled + hole-filled intermediate, computed
// from the x tile in LDS. xr0/xc0 are the (unclamped) x coords of LDS slot (0,0).
__device__ __forceinline__ float res_val(const float* __restrict__ sx,
                                         int r, int c, int xr0, int xc0) {
    const int ch = c >> 1;          // x column for both parities
    const int rr = r  - xr0;        // LDS row slot
    const int cc = ch - xc0;        // LDS col slot
    if (((r ^ c) & 1) == 0) {
        // filled checkerboard position: direct copy of x
        return sx[rr * XTW + cc];
    }
    if ((r & 1) == 0) {
        // even row, odd col hole; last odd column never filled
        if (c == RES_W - 1) return 0.0f;
        const int rm = (r == 0) ? 1 : (r - 1);   // reflect row -1 -> 1
        return 0.25f * (sx[(rm - xr0) * XTW + cc] +
                        sx[(rr + 1)   * XTW + cc] +
                        sx[rr * XTW + cc] +
                        sx[rr * XTW + cc + 1]);
    } else {
        // odd row, even col hole; last row never filled
        if (r == RES_H - 1) return 0.0f;
        const int cm = (c == 0) ? (0 - xc0) : (cc - 1);  // reflect col -1 -> 1 => x col 0
        return 0.25f * (sx[(rr - 1) * XTW + cc] +
                        sx[(rr + 1) * XTW + cc] +
                        sx[rr * XTW + cm] +
                        sx[rr * XTW + cc]);
    }
}

__global__ __launch_bounds__(256)
void UpsampleTSDConv_fused_kernel(const float* __restrict__ x,
                                  float* __restrict__ out) {
    const int tid = threadIdx.x;
    const int j0  = blockIdx.x * TOW;   // output col tile base (flipped domain)
    const int i0  = blockIdx.y * TOH;   // output row tile base
    const int b   = blockIdx.z;

    // res-space / x-space tile origins
    const int kr0 = i0 >> 1;                        // first res row touched + 1
    const int jj0 = OUT_WW - TOW - j0;              // unflipped col base
    const int cl0 = jj0 >> 1;                       // res col tile base
    const int xr0 = kr0 - 2;                        // x tile row origin (unclamped)
    const int xc0 = (cl0 >> 1) - 2;                 // x tile col origin (unclamped)

    __shared__ float sx[XTH * XTW];                 // 12x20 floats = 960 B

    const float* xb = x + (size_t)b * (H_IN * W_IN);

    // ---- Stage x tile into LDS via CDNA5 async global->LDS path ----
    if (tid < XTH * XTW) {
        const int lrow = tid / XTW;
        const int lcol = tid - lrow * XTW;
        int gr = xr0 + lrow; gr = gr < 0 ? 0 : (gr > H_IN - 1 ? H_IN - 1 : gr);
        int gc = xc0 + lcol; gc = gc < 0 ? 0 : (gc > W_IN - 1 ? W_IN - 1 : gc);
        const float* gsrc = xb + gr * W_IN + gc;
        // Low 32 bits of a generic pointer to LDS are the LDS byte offset.
        unsigned lds_off = (unsigned)(size_t)(&sx[tid]);
        asm volatile("global_load_async_to_lds_b32 %0, %1, off"
                     :
                     : "v"(lds_off), "v"(gsrc)
                     : "memory");
    }
    asm volatile("s_wait_asynccnt 0x0" ::: "memory");
    __syncthreads();

    // ---- Compute 4 consecutive output pixels per thread ----
    const int lr  = tid >> 4;           // 0..15 : output row within tile
    const int lcg = tid & 15;           // 0..15 : 4-col group within tile
    const int i   = i0 + lr;
    const int jb  = j0 + (lcg << 2);

    // Row interpolation (half-pixel bilinear, scale 2): weights {0.25, 0.75}
    const int k = i >> 1;
    int r0, r1; float wr0, wr1;
    if ((i & 1) == 0) { r0 = (k > 0) ? k - 1 : 0; r1 = k;  wr0 = 0.25f; wr1 = 0.75f; }
    else              { r0 = k; r1 = (k < RES_H - 1) ? k + 1 : RES_H - 1; wr0 = 0.75f; wr1 = 0.25f; }

    // Column positions in unflipped space: jj = 639-jb .. 636-jb, jj = 2L+1 down to 2L-2
    const int JJ = OUT_WW - 1 - jb;     // odd
    const int L  = JJ >> 1;             // in [1, 319]
    int c0 = L - 2; c0 = c0 < 0 ? 0 : c0;
    int c3 = L + 1; c3 = c3 > RES_W - 1 ? RES_W - 1 : c3;
    const int cols[4] = { c0, L - 1, L, c3 };

    // Row-blended res values for the 4 distinct columns
    float t[4];
#pragma unroll
    for (int q = 0; q < 4; ++q) {
        t[q] = wr0 * res_val(sx, r0, cols[q], xr0, xc0)
             + wr1 * res_val(sx, r1, cols[q], xr0, xc0);
    }

    // Column interpolation + LR flip baked in:
    //   j=jb+0: jj=2L+1 -> 0.75 t(L)   + 0.25 t(L+1)
    //   j=jb+1: jj=2L   -> 0.25 t(L-1) + 0.75 t(L)
    //   j=jb+2: jj=2L-1 -> 0.75 t(L-1) + 0.25 t(L)
    //   j=jb+3: jj=2L-2 -> 0.25 t(L-2) + 0.75 t(L-1)
    float4 o;
    o.x = 0.75f * t[2] + 0.25f * t[3];
    o.y = 0.25f * t[1] + 0.75f * t[2];
    o.z = 0.75f * t[1] + 0.25f * t[2];
    o.w = 0.25f * t[0] + 0.75f * t[1];

    *(float4*)(out + ((size_t)b * OUT_HH + i) * OUT_WW + jb) = o;
}

extern "C" void kernel_launch(void* const* d_in, const int* in_sizes, int n_in,
                              void* d_out, int out_size, void* d_ws, size_t ws_size,
                              hipStream_t stream) {
    const float* x = (const float*)d_in[0];
    float* out = (float*)d_out;
    const int B = in_sizes[0] / (H_IN * W_IN);   // 256

    dim3 grid(OUT_WW / TOW, OUT_HH / TOH, B);    // 10 x 20 x 256
    dim3 block(256);
    hipLaunchKernelGGL(UpsampleTSDConv_fused_kernel, grid, block, 0, stream, x, out);
}